// MoCTopKExperts_18176301596933
// MI455X (gfx1250) — compile-verified
//
#include <hip/hip_runtime.h>
#include <hip/hip_bf16.h>

// MoE top-2 experts, fp32 end-to-end, CDNA5 (gfx1250, wave32) WMMA f32 16x16x4.
// Workspace requirement: ~295 MB.

#define N_TOK   16384
#define D_MODEL 768
#define N_EXP   8
#define TOPK    2
#define CAP     5120
#define HFF     2048

typedef __attribute__((ext_vector_type(2))) float v2f;
typedef __attribute__((ext_vector_type(8))) float v8f;
typedef int v4i __attribute__((vector_size(16)));  // matches builtin param type

__device__ __forceinline__ v8f wmma4(v2f a, v2f b, v8f c) {
  // D = A(16x4) * B(4x16) + C, f32, wave32.
  return __builtin_amdgcn_wmma_f32_16x16x4_f32(false, a, false, b, (short)0, c,
                                               false, false);
}

// ---- gfx1250 async global->LDS path (probe via __has_builtin) --------------
#if defined(__has_builtin)
#if __has_builtin(__builtin_amdgcn_global_load_async_to_lds_b128)
#define HAVE_ASYNC_LDS 1
#endif
#if __has_builtin(__builtin_amdgcn_s_wait_asynccnt)
#define HAVE_WAIT_ASYNC 1
#endif
#endif

#if HAVE_ASYNC_LDS
#define ASYNC16(gp, lp)                                                        \
  __builtin_amdgcn_global_load_async_to_lds_b128(                              \
      (__attribute__((address_space(1))) v4i*)(void*)(gp),                     \
      (__attribute__((address_space(3))) v4i*)(void*)(lp), 0, 0)
#endif
#if HAVE_WAIT_ASYNC
#define WAIT_ASYNC(n) __builtin_amdgcn_s_wait_asynccnt(n)
#else
#define WAIT_ASYNC(n) asm volatile("s_wait_asynccnt " #n ::: "memory")
#endif

// ---------------------------------------------------------------- router ----
__global__ __launch_bounds__(128) void moe_router(const float* __restrict__ x,
                                                  const float* __restrict__ gate_w,
                                                  float* __restrict__ cw,
                                                  int* __restrict__ top_i) {
  __shared__ __align__(16) float gws[N_EXP * D_MODEL];
  const int tid = threadIdx.x;
  for (int i = tid; i < N_EXP * D_MODEL / 4; i += 128)
    ((float4*)gws)[i] = ((const float4*)gate_w)[i];
  __syncthreads();

  const int wid = tid >> 5, ln = tid & 31;
  const int n = blockIdx.x * 4 + wid;
  const float* xr = x + (size_t)n * D_MODEL;

  float acc[N_EXP];
#pragma unroll
  for (int e = 0; e < N_EXP; ++e) acc[e] = 0.0f;

  for (int d = ln * 4; d < D_MODEL; d += 128) {
    const float4 xv = *(const float4*)(xr + d);
#pragma unroll
    for (int e = 0; e < N_EXP; ++e) {
      const float4 g = *(const float4*)(gws + e * D_MODEL + d);
      acc[e] += xv.x * g.x + xv.y * g.y + xv.z * g.z + xv.w * g.w;
    }
  }
#pragma unroll
  for (int e = 0; e < N_EXP; ++e)
    for (int off = 16; off; off >>= 1) acc[e] += __shfl_xor(acc[e], off, 32);

  if (ln == 0) {
    int i1 = 0; float m1 = acc[0];
#pragma unroll
    for (int e = 1; e < N_EXP; ++e)
      if (acc[e] > m1) { m1 = acc[e]; i1 = e; }
    int i2 = -1; float m2 = -3.4e38f;
#pragma unroll
    for (int e = 0; e < N_EXP; ++e)
      if (e != i1 && acc[e] > m2) { m2 = acc[e]; i2 = e; }
    const float t = __expf(m2 - m1);
    const float inv = 1.0f / (1.0f + t);
    cw[2 * n + 0] = inv;
    cw[2 * n + 1] = t * inv;
    top_i[2 * n + 0] = i1;
    top_i[2 * n + 1] = i2;
  }
}

// ------------------------------------------------------------- dispatch -----
__global__ __launch_bounds__(256) void moe_scan(const int* __restrict__ top_i,
                                                float* __restrict__ cw,
                                                int* __restrict__ didx,
                                                int* __restrict__ slot_tok,
                                                int* __restrict__ filled) {
  __shared__ int cnt[256 * N_EXP];
  const int t = threadIdx.x;
  const int base = t * (N_TOK * TOPK / 256);  // 128 entries per thread
#pragma unroll
  for (int e = 0; e < N_EXP; ++e) cnt[t * N_EXP + e] = 0;
  for (int i = 0; i < 128; ++i) cnt[t * N_EXP + top_i[base + i]]++;
  __syncthreads();
  if (t < N_EXP) {  // serial exclusive scan per expert
    int run = 0;
    for (int q = 0; q < 256; ++q) {
      const int v = cnt[q * N_EXP + t];
      cnt[q * N_EXP + t] = run;
      run += v;
    }
    filled[t] = run < CAP ? run : CAP;
  }
  __syncthreads();
  for (int i = 0; i < 128; ++i) {
    const int j = base + i;
    const int e = top_i[j];
    const int pos = cnt[t * N_EXP + e]++;
    const bool keep = pos < CAP;
    const int slot = e * CAP + (keep ? pos : (CAP - 1));
    didx[j] = slot;
    if (keep) slot_tok[slot] = j >> 1;
    else      cw[j] = 0.0f;
  }
}

// ------------------------------------------------------ gather + RMSNorm ----
__global__ __launch_bounds__(256) void moe_gather_norm(
    const float* __restrict__ x, const float* __restrict__ norm_w,
    const int* __restrict__ top_i, const int* __restrict__ didx,
    const int* __restrict__ slot_tok, float* __restrict__ z) {
  const int j = blockIdx.x;
  const int s = didx[j];
  const int n = j >> 1;
  if (slot_tok[s] != n) return;  // dropped entry
  const int e = top_i[j];
  const int t = threadIdx.x;
  const float* xr = x + (size_t)n * D_MODEL;
  const float v0 = xr[t], v1 = xr[t + 256], v2 = xr[t + 512];
  float ss = v0 * v0 + v1 * v1 + v2 * v2;
  for (int off = 16; off; off >>= 1) ss += __shfl_xor(ss, off, 32);
  __shared__ float wsum[8];
  const int wid = t >> 5, ln = t & 31;
  if (ln == 0) wsum[wid] = ss;
  __syncthreads();
  float tot = 0.0f;
#pragma unroll
  for (int q = 0; q < 8; ++q) tot += wsum[q];
  const float scale = 2.0f * rsqrtf(4.0f * tot / (float)D_MODEL + 1e-6f);
  const float* nw = norm_w + (size_t)e * D_MODEL;
  float* zr = z + (size_t)s * D_MODEL;
  zr[t]       = v0 * scale * nw[t];
  zr[t + 256] = v1 * scale * nw[t + 256];
  zr[t + 512] = v2 * scale * nw[t + 512];
}

// ------------------------------------------------------------- GEMM core ----
// Row-major LDS tiles, stride 20 dwords: fragment (K,K+1) pair = one b64 read,
// bank-conflict-free for both half-wave K offsets; 16B-aligned fills.
#define KT   16
#define LSTR 20                 // 16 + 4 pad dwords
#define STG  (128 * LSTR)       // floats per stage buffer

// GEMM1: hs = silu(z.w13g^T) * (z.w13u^T).  Block M=128, N=64 per g/u half.
__global__ __launch_bounds__(256) void moe_gemm1(
    const float* __restrict__ z_e, const float* __restrict__ w13_e,
    float* __restrict__ hs, const int* __restrict__ pfilled) {
  const int m0 = blockIdx.y * 128;
  if (m0 >= *pfilled) return;
  const int hb = blockIdx.x * 64;

#if HAVE_ASYNC_LDS
  __shared__ float At[3][STG];
  __shared__ float Bt[3][STG];
#else
  __shared__ float At[2][STG];
  __shared__ float Bt[2][STG];
#endif

  const int tid = threadIdx.x;
  const int ln = tid & 31, wid = tid >> 5;
  const int l16 = ln & 15, hi = ln >> 4;
  const int wm = wid >> 1, wn = wid & 1;
  const int mA0 = wm * 32 + l16;

  // two 16B chunks per thread: chunk0 covers rows 0..63, chunk1 rows 64..127
  const int row0 = tid >> 2, kk0 = (tid & 3) << 2;
  const int row1 = row0 + 64, kk1 = kk0;
  const int hrow0 = hb + row0;                 // g rows
  const int hrow1 = HFF + hb + (row1 - 64);    // u rows

  v8f cg[2][2], cu[2][2];
#pragma unroll
  for (int i = 0; i < 2; ++i)
#pragma unroll
    for (int j = 0; j < 2; ++j)
#pragma unroll
      for (int q = 0; q < 8; ++q) { cg[i][j][q] = 0.0f; cu[i][j][q] = 0.0f; }

  const int S = D_MODEL / KT;  // 48 stages

  auto compute = [&](const float* A, const float* B) {
#pragma unroll
    for (int kl = 0; kl < KT; kl += 4) {
      const int kr = kl + 2 * hi;
      v2f a[2], bg[2], bu[2];
#pragma unroll
      for (int i = 0; i < 2; ++i) {
        const float2 f = *(const float2*)(A + (mA0 + i * 16) * LSTR + kr);
        a[i].x = f.x; a[i].y = f.y;
      }
#pragma unroll
      for (int j = 0; j < 2; ++j) {
        const int ng = wn * 32 + j * 16 + l16;
        const float2 fg = *(const float2*)(B + ng * LSTR + kr);
        bg[j].x = fg.x; bg[j].y = fg.y;
        const float2 fu = *(const float2*)(B + (64 + ng) * LSTR + kr);
        bu[j].x = fu.x; bu[j].y = fu.y;
      }
#pragma unroll
      for (int i = 0; i < 2; ++i)
#pragma unroll
        for (int j = 0; j < 2; ++j) {
          cg[i][j] = wmma4(a[i], bg[j], cg[i][j]);
          cu[i][j] = wmma4(a[i], bu[j], cu[i][j]);
        }
    }
  };

#if HAVE_ASYNC_LDS
  auto issue = [&](int k0, int b) {
    ASYNC16(z_e + (size_t)(m0 + row0) * D_MODEL + k0 + kk0, &At[b][row0 * LSTR + kk0]);
    ASYNC16(z_e + (size_t)(m0 + row1) * D_MODEL + k0 + kk1, &At[b][row1 * LSTR + kk1]);
    ASYNC16(w13_e + (size_t)hrow0 * D_MODEL + k0 + kk0, &Bt[b][row0 * LSTR + kk0]);
    ASYNC16(w13_e + (size_t)hrow1 * D_MODEL + k0 + kk1, &Bt[b][row1 * LSTR + kk1]);
  };
  issue(0, 0);
  issue(KT, 1);
  for (int s = 0; s < S; ++s) {
    if (s + 1 < S) { WAIT_ASYNC(4); } else { WAIT_ASYNC(0); }
    __syncthreads();
    const int b = s % 3;
    compute(At[b], Bt[b]);
    if (s + 2 < S) issue((s + 2) * KT, (s + 2) % 3);  // after barrier: WAR-safe
  }
#else
  float4 ra0, ra1, rb0, rb1;
  auto gload = [&](int k0) {
    ra0 = *(const float4*)(z_e + (size_t)(m0 + row0) * D_MODEL + k0 + kk0);
    ra1 = *(const float4*)(z_e + (size_t)(m0 + row1) * D_MODEL + k0 + kk1);
    rb0 = *(const float4*)(w13_e + (size_t)hrow0 * D_MODEL + k0 + kk0);
    rb1 = *(const float4*)(w13_e + (size_t)hrow1 * D_MODEL + k0 + kk1);
  };
  gload(0);
  for (int s = 0; s < S; ++s) {
    const int b = s & 1;
    *(float4*)&At[b][row0 * LSTR + kk0] = ra0;
    *(float4*)&At[b][row1 * LSTR + kk1] = ra1;
    *(float4*)&Bt[b][row0 * LSTR + kk0] = rb0;
    *(float4*)&Bt[b][row1 * LSTR + kk1] = rb1;
    if (s + 1 < S) gload((s + 1) * KT);
    __syncthreads();
    compute(At[b], Bt[b]);
  }
#endif

  // epilogue: hs = silu(g) * u
#pragma unroll
  for (int i = 0; i < 2; ++i) {
    const int rbase = m0 + wm * 32 + i * 16 + hi * 8;
#pragma unroll
    for (int j = 0; j < 2; ++j) {
      const int h = hb + wn * 32 + j * 16 + l16;
#pragma unroll
      for (int v = 0; v < 8; ++v) {
        const float g = cg[i][j][v], u = cu[i][j][v];
        const float sg = g / (1.0f + __expf(-g));
        hs[(size_t)(rbase + v) * HFF + h] = sg * u;
      }
    }
  }
}

// GEMM2: y = x[slot_token] + hs . w2^T.  Block M=128, N=128, K=2048.
__global__ __launch_bounds__(256) void moe_gemm2(
    const float* __restrict__ hs, const float* __restrict__ w2_e,
    const float* __restrict__ x, const int* __restrict__ slot_tok_e,
    float* __restrict__ y_e, const int* __restrict__ pfilled) {
  const int m0 = blockIdx.y * 128;
  if (m0 >= *pfilled) return;
  const int db = blockIdx.x * 128;

#if HAVE_ASYNC_LDS
  __shared__ float At[3][STG];
  __shared__ float Bt[3][STG];
#else
  __shared__ float At[2][STG];
  __shared__ float Bt[2][STG];
#endif

  const int tid = threadIdx.x;
  const int ln = tid & 31, wid = tid >> 5;
  const int l16 = ln & 15, hi = ln >> 4;
  const int wm = wid >> 1, wn = wid & 1;
  const int mA0 = wm * 32 + l16;

  const int row0 = tid >> 2, kk0 = (tid & 3) << 2;
  const int row1 = row0 + 64, kk1 = kk0;

  v8f c[2][4];
#pragma unroll
  for (int i = 0; i < 2; ++i)
#pragma unroll
    for (int j = 0; j < 4; ++j)
#pragma unroll
      for (int q = 0; q < 8; ++q) c[i][j][q] = 0.0f;

  const int S = HFF / KT;  // 128 stages

  auto compute = [&](const float* A, const float* B) {
#pragma unroll
    for (int kl = 0; kl < KT; kl += 4) {
      const int kr = kl + 2 * hi;
      v2f a[2], b[4];
#pragma unroll
      for (int i = 0; i < 2; ++i) {
        const float2 f = *(const float2*)(A + (mA0 + i * 16) * LSTR + kr);
        a[i].x = f.x; a[i].y = f.y;
      }
#pragma unroll
      for (int j = 0; j < 4; ++j) {
        const float2 f = *(const float2*)(B + (wn * 64 + j * 16 + l16) * LSTR + kr);
        b[j].x = f.x; b[j].y = f.y;
      }
#pragma unroll
      for (int i = 0; i < 2; ++i)
#pragma unroll
        for (int j = 0; j < 4; ++j) c[i][j] = wmma4(a[i], b[j], c[i][j]);
    }
  };

#if HAVE_ASYNC_LDS
  auto issue = [&](int k0, int b) {
    ASYNC16(hs + (size_t)(m0 + row0) * HFF + k0 + kk0, &At[b][row0 * LSTR + kk0]);
    ASYNC16(hs + (size_t)(m0 + row1) * HFF + k0 + kk1, &At[b][row1 * LSTR + kk1]);
    ASYNC16(w2_e + (size_t)(db + row0) * HFF + k0 + kk0, &Bt[b][row0 * LSTR + kk0]);
    ASYNC16(w2_e + (size_t)(db + row1) * HFF + k0 + kk1, &Bt[b][row1 * LSTR + kk1]);
  };
  issue(0, 0);
  issue(KT, 1);
  for (int s = 0; s < S; ++s) {
    if (s + 1 < S) { WAIT_ASYNC(4); } else { WAIT_ASYNC(0); }
    __syncthreads();
    const int b = s % 3;
    compute(At[b], Bt[b]);
    if (s + 2 < S) issue((s + 2) * KT, (s + 2) % 3);
  }
#else
  float4 ra0, ra1, rb0, rb1;
  auto gload = [&](int k0) {
    ra0 = *(const float4*)(hs + (size_t)(m0 + row0) * HFF + k0 + kk0);
    ra1 = *(const float4*)(hs + (size_t)(m0 + row1) * HFF + k0 + kk1);
    rb0 = *(const float4*)(w2_e + (size_t)(db + row0) * HFF + k0 + kk0);
    rb1 = *(const float4*)(w2_e + (size_t)(db + row1) * HFF + k0 + kk1);
  };
  gload(0);
  for (int s = 0; s < S; ++s) {
    const int b = s & 1;
    *(float4*)&At[b][row0 * LSTR + kk0] = ra0;
    *(float4*)&At[b][row1 * LSTR + kk1] = ra1;
    *(float4*)&Bt[b][row0 * LSTR + kk0] = rb0;
    *(float4*)&Bt[b][row1 * LSTR + kk1] = rb1;
    if (s + 1 < S) gload((s + 1) * KT);
    __syncthreads();
    compute(At[b], Bt[b]);
  }
#endif

  // epilogue: residual add of gathered token row, write y
#pragma unroll
  for (int i = 0; i < 2; ++i) {
    const int rbase = m0 + wm * 32 + i * 16 + hi * 8;
    int toks[8];
#pragma unroll
    for (int v = 0; v < 8; ++v) {
      const int tk = slot_tok_e[rbase + v];
      toks[v] = ((unsigned)tk < (unsigned)N_TOK) ? tk : 0;
    }
#pragma unroll
    for (int j = 0; j < 4; ++j) {
      const int dcol = db + wn * 64 + j * 16 + l16;
#pragma unroll
      for (int v = 0; v < 8; ++v) {
        const float res = x[(size_t)toks[v] * D_MODEL + dcol];
        y_e[(size_t)(rbase + v) * D_MODEL + dcol] = res + c[i][j][v];
      }
    }
  }
}

// -------------------------------------------------------------- combine -----
__global__ __launch_bounds__(192) void moe_combine(const float* __restrict__ y,
                                                   const int* __restrict__ didx,
                                                   const float* __restrict__ cw,
                                                   float* __restrict__ out) {
  const int n = blockIdx.x, t = threadIdx.x;  // 192 threads * float4 = 768
  const int s0 = didx[2 * n], s1 = didx[2 * n + 1];
  const float w0 = cw[2 * n], w1 = cw[2 * n + 1];
  const float4 a = ((const float4*)(y + (size_t)s0 * D_MODEL))[t];
  const float4 b = ((const float4*)(y + (size_t)s1 * D_MODEL))[t];
  float4 o;
  o.x = w0 * a.x + w1 * b.x;
  o.y = w0 * a.y + w1 * b.y;
  o.z = w0 * a.z + w1 * b.z;
  o.w = w0 * a.w + w1 * b.w;
  ((float4*)(out + (size_t)n * D_MODEL))[t] = o;
}

// ---------------------------------------------------------------------------
extern "C" void kernel_launch(void* const* d_in, const int* in_sizes, int n_in,
                              void* d_out, int out_size, void* d_ws,
                              size_t ws_size, hipStream_t stream) {
  (void)in_sizes; (void)n_in; (void)out_size; (void)ws_size;
  const float* x      = (const float*)d_in[0];  // [4,4096,768]
  const float* gate_w = (const float*)d_in[1];  // [8,768]
  const float* w13    = (const float*)d_in[2];  // [8,4096,768]
  const float* w2     = (const float*)d_in[3];  // [8,768,2048]
  const float* norm_w = (const float*)d_in[4];  // [8,768]
  float* out = (float*)d_out;

  char* p = (char*)d_ws;
  auto take = [&](size_t bytes) {
    char* q = p;
    p += (bytes + 255) & ~(size_t)255;
    return q;
  };
  float* cw       = (float*)take(sizeof(float) * N_TOK * TOPK);
  int*   top_i    = (int*)take(sizeof(int) * N_TOK * TOPK);
  int*   didx     = (int*)take(sizeof(int) * N_TOK * TOPK);
  int*   slot_tok = (int*)take(sizeof(int) * N_EXP * CAP);
  int*   filled   = (int*)take(sizeof(int) * N_EXP);
  float* z        = (float*)take(sizeof(float) * (size_t)N_EXP * CAP * D_MODEL);
  float* hs       = (float*)take(sizeof(float) * (size_t)CAP * HFF);  // reused
  float* y        = (float*)take(sizeof(float) * (size_t)N_EXP * CAP * D_MODEL);

  moe_router<<<N_TOK / 4, 128, 0, stream>>>(x, gate_w, cw, top_i);
  moe_scan<<<1, 256, 0, stream>>>(top_i, cw, didx, slot_tok, filled);
  moe_gather_norm<<<N_TOK * TOPK, 256, 0, stream>>>(x, norm_w, top_i, didx,
                                                    slot_tok, z);
  for (int e = 0; e < N_EXP; ++e) {
    moe_gemm1<<<dim3(HFF / 64, CAP / 128), 256, 0, stream>>>(
        z + (size_t)e * CAP * D_MODEL, w13 + (size_t)e * 2 * HFF * D_MODEL, hs,
        filled + e);
    moe_gemm2<<<dim3(D_MODEL / 128, CAP / 128), 256, 0, stream>>>(
        hs, w2 + (size_t)e * D_MODEL * HFF, x, slot_tok + (size_t)e * CAP,
        y + (size_t)e * CAP * D_MODEL, filled + e);
  }
  moe_combine<<<N_TOK, 192, 0, stream>>>(y, didx, cw, out);
}